// TopologyLoss_618475291392
// MI455X (gfx1250) — compile-verified
//
#include <hip/hip_runtime.h>
#include <hip/hip_bf16.h>

// Problem constants from the reference
#define TL_B   8
#define TL_C   4
#define TL_HW  (512 * 512)
#define TL_K   100          // 2 * MAX_PAIRS
#define TL_NTERMS (TL_B * TL_K)   // 800
#define TL_CRACK 1

typedef __attribute__((ext_vector_type(2))) float v2f;
typedef __attribute__((ext_vector_type(8))) float v8f;

// Single-workgroup kernel: gather-first softmax for the 800 loss terms,
// then a full-f32 WMMA-based tree reduction of the 1024-slot contribution
// array (padded with zeros) down to the scalar loss.
__global__ __launch_bounds__(1024) void TopologyLoss_618475291392_kernel(
    const float* __restrict__ logits,      // [B, C, H*W]
    const int*   __restrict__ term_idx,    // [B, K]
    const float* __restrict__ term_tgt,    // [B, K]
    const float* __restrict__ term_valid,  // [B, K]
    const float* __restrict__ term_count,  // [B]
    float* __restrict__ out)               // [1]
{
    __shared__ float contrib[1024];

    const int tid = threadIdx.x;

    float c = 0.0f;
    if (tid < TL_NTERMS) {
        const int b   = tid / TL_K;
        const int idx = term_idx[tid];
        const float* base = logits + (size_t)b * TL_C * TL_HW + (size_t)idx;
        // 4-class softmax at this single pixel (scattered L2 reads).
        const float l0 = base[0 * TL_HW];
        const float l1 = base[1 * TL_HW];
        const float l2 = base[2 * TL_HW];
        const float l3 = base[3 * TL_HW];
        const float m  = fmaxf(fmaxf(l0, l1), fmaxf(l2, l3));
        const float e0 = expf(l0 - m);
        const float e1 = expf(l1 - m);   // crack class = 1
        const float e2 = expf(l2 - m);
        const float e3 = expf(l3 - m);
        const float p  = e1 / (e0 + e1 + e2 + e3);
        const float d  = p - term_tgt[tid];
        // fold per-image 1/count and batch 1/B into the term
        c = term_valid[tid] * d * d / (term_count[b] * (float)TL_B);
    }
    contrib[tid] = c;
    __syncthreads();

    // Wave 0 (lanes 0..31, EXEC all-ones as required by WMMA) reduces all
    // 1024 contributions with V_WMMA_F32_16X16X4_F32:
    //   A = ones(16x4)  ->  D[m][n] = sum_k B[k][n] + C[m][n]
    // 16 accumulating WMMAs consume 16 * 64 = 1024 values; row M=0 of the
    // accumulator (acc[0], lanes 0..15) then holds the 16 column sums.
    if (tid < 32) {
        v2f a;                 // all-ones A matrix: every lane's 2 A slots = 1.0
        a[0] = 1.0f;
        a[1] = 1.0f;
        v8f acc = {};          // C/D accumulator starts at zero

#pragma unroll
        for (int chunk = 0; chunk < 16; ++chunk) {
            v2f bb;            // 64 contributions per chunk, 2 per lane
            bb[0] = contrib[chunk * 64 + tid * 2 + 0];
            bb[1] = contrib[chunk * 64 + tid * 2 + 1];
            // 8 args: (neg_a, A, neg_b, B, c_mod, C, reuse_a, reuse_b)
            acc = __builtin_amdgcn_wmma_f32_16x16x4_f32(
                /*neg_a=*/false, a, /*neg_b=*/false, bb,
                /*c_mod=*/(short)0, acc, /*reuse_a=*/false, /*reuse_b=*/false);
        }

        // Row M=0 lives in acc element 0, lanes 0..15 (N = lane).
        float v = (tid < 16) ? acc[0] : 0.0f;
        v += __shfl_xor(v, 8, 32);
        v += __shfl_xor(v, 4, 32);
        v += __shfl_xor(v, 2, 32);
        v += __shfl_xor(v, 1, 32);
        if (tid == 0) out[0] = v;
    }
}

extern "C" void kernel_launch(void* const* d_in, const int* in_sizes, int n_in,
                              void* d_out, int out_size, void* d_ws, size_t ws_size,
                              hipStream_t stream) {
    const float* logits     = (const float*)d_in[0];
    // d_in[1] = masks: unused by the loss (term construction happened host-side)
    const int*   term_idx   = (const int*)  d_in[2];
    const float* term_tgt   = (const float*)d_in[3];
    const float* term_valid = (const float*)d_in[4];
    const float* term_count = (const float*)d_in[5];
    float* out = (float*)d_out;

    hipLaunchKernelGGL(TopologyLoss_618475291392_kernel,
                       dim3(1), dim3(1024), 0, stream,
                       logits, term_idx, term_tgt, term_valid, term_count, out);
}